// FourierIntegralLayer_50680614092830
// MI455X (gfx1250) — compile-verified
//
#include <hip/hip_runtime.h>
#include <math.h>

// Problem constants (from reference)
#define BB   16          // batch
#define NN   4096        // spatial grid
#define DD   64          // channels
#define KM   32          // truncated modes
#define PP   8           // params
#define HH   128         // mlp hidden
#define KD   (KM*DD*DD)  // 131072 = W2 output dim

typedef __attribute__((ext_vector_type(2))) float v2f;
typedef __attribute__((ext_vector_type(8))) float v8f;

// ---------------------------------------------------------------------------
// Kernel 1: trig table  T[n,k] = (cos, sin)(2*pi*k*n/N), 4096*32 entries (1MB)
// Shared by forward DFT (e^{-i..}) and inverse DFT (e^{+i..}).
// Use (k*n mod N) for phase accuracy.
// ---------------------------------------------------------------------------
__global__ void trig_kernel(float* __restrict__ cosT, float* __restrict__ sinT) {
    int idx = blockIdx.x * blockDim.x + threadIdx.x;          // idx = n*KM + k
    int n = idx / KM, k = idx % KM;
    int m = (n * k) & (NN - 1);
    float ph = (float)m * (6.28318530717958647692f / (float)NN);
    float s, c;
    __sincosf(ph, &s, &c);
    cosT[idx] = c;
    sinT[idx] = s;
}

// ---------------------------------------------------------------------------
// Kernel 2: truncated forward DFT.  X[b,k,d] = sum_n f[b,n,d]*(cos - i*sin)
// block = (64 d-lanes, 4 modes); grid = (KM/4, B). f row loads are 256B
// coalesced; trig loads are wave-broadcast; f stays L2-resident.
// ---------------------------------------------------------------------------
__global__ __launch_bounds__(256) void dft_kernel(
    const float* __restrict__ f, const float* __restrict__ cosT,
    const float* __restrict__ sinT, float* __restrict__ Xr, float* __restrict__ Xi) {
    int d = threadIdx.x;
    int k = blockIdx.x * 4 + threadIdx.y;
    int b = blockIdx.y;
    float aR = 0.f, aI = 0.f;
    const float* fb = f + (size_t)b * NN * DD + d;
    for (int n = 0; n < NN; ++n) {
        float fv = fb[(size_t)n * DD];
        float c = cosT[n * KM + k];
        float s = sinT[n * KM + k];
        aR = fmaf(fv, c, aR);
        aI = fmaf(-fv, s, aI);
    }
    Xr[(b * KM + k) * DD + d] = aR;
    Xi[(b * KM + k) * DD + d] = aI;
}

// ---------------------------------------------------------------------------
// Kernel 3: MLP hidden layer with swish.  h[b,h] = swish(p @ W1 + b1)
// grid = (B, 2) [y: 0 = real branch, 1 = imag branch], block = 128 (h)
// ---------------------------------------------------------------------------
__global__ __launch_bounds__(128) void mlp_kernel(
    const float* __restrict__ params,
    const float* __restrict__ W1r, const float* __restrict__ b1r,
    const float* __restrict__ W1c, const float* __restrict__ b1c,
    float* __restrict__ hr, float* __restrict__ hc) {
    int b = blockIdx.x;
    int h = threadIdx.x;
    const float* W1 = blockIdx.y ? W1c : W1r;
    const float* b1 = blockIdx.y ? b1c : b1r;
    float*     hout = blockIdx.y ? hc  : hr;
    float acc = b1[h];
#pragma unroll
    for (int p = 0; p < PP; ++p)
        acc = fmaf(params[b * PP + p], W1[p * HH + h], acc);
    hout[b * HH + h] = acc / (1.f + __expf(-acc));   // swish
}

// ---------------------------------------------------------------------------
// Kernel 4: THE bandwidth-dominant GEMM, done with native f32 WMMA.
//   kout[b, col] = sum_h hmat[b,h] * W2[h,col] + b2[col]
//   M=16 (batch) x K=128 (hidden) x N=131072 (k*d*d columns)
// One wave owns one 16-column tile; 32 chained V_WMMA_F32_16X16X4_F32 over K.
// Operand layouts per CDNA5 ISA 7.12.2 (32-bit):
//   A 16x4 : v0 = A[lm][k0+2*half], v1 = A[lm][k0+2*half+1]
//   B 4x16 : v0 = B[k0+2*half][lm], v1 = B[k0+2*half+1][lm]
//   C/D    : acc[r] = D[r + 8*half][lm]
// W2 is streamed once from HBM (134 MB total) => this stage sets the runtime.
// grid = (KD/64, 2), block = 128 (4 waves -> 64 columns per block)
// ---------------------------------------------------------------------------
__global__ __launch_bounds__(128) void gemm_wmma_kernel(
    const float* __restrict__ hr, const float* __restrict__ hc,
    const float* __restrict__ W2r, const float* __restrict__ b2r,
    const float* __restrict__ W2c, const float* __restrict__ b2c,
    float* __restrict__ kr, float* __restrict__ kc) {
    const float* hm = blockIdx.y ? hc  : hr;
    const float* W2 = blockIdx.y ? W2c : W2r;
    const float* b2 = blockIdx.y ? b2c : b2r;
    float*     kout = blockIdx.y ? kc  : kr;

    int lane = threadIdx.x & 31;
    int wave = threadIdx.x >> 5;
    int col0 = blockIdx.x * 64 + wave * 16;
    int half = lane >> 4;        // 0: lanes 0-15, 1: lanes 16-31
    int lm   = lane & 15;

    v8f acc = {};
#pragma unroll
    for (int kk = 0; kk < HH; kk += 4) {
        // A fragment (hidden activations, L1-resident)
        v2f av = { hm[lm * HH + kk + 2 * half],
                   hm[lm * HH + kk + 2 * half + 1] };
        // B fragment (streamed W2): two 64B segments per load, adjacent waves
        // cover the sibling 64B sectors of each 128B line.
        size_t rb = (size_t)(kk + 2 * half) * KD + (size_t)col0 + lm;
        v2f bv = { W2[rb], W2[rb + KD] };
        acc = __builtin_amdgcn_wmma_f32_16x16x4_f32(
            false, av, false, bv, (short)0, acc, false, false);
    }

    float bias = b2[col0 + lm];          // bias depends only on column
#pragma unroll
    for (int r = 0; r < 8; ++r) {
        int row = r + 8 * half;          // batch index
        kout[(size_t)row * KD + (size_t)col0 + lm] = acc[r] + bias;
    }
}

// ---------------------------------------------------------------------------
// Kernel 5: per-(b,k) complex matvec  y = (kr + i*kc) @ (Xr + i*Xi), * D
// grid = (KM, B), block = 64 (output row i). X slice staged in LDS;
// kernel rows read as float4 (256B per thread, L2-resident).
// ---------------------------------------------------------------------------
__global__ __launch_bounds__(64) void matvec_kernel(
    const float* __restrict__ kr, const float* __restrict__ kc,
    const float* __restrict__ Xr, const float* __restrict__ Xi,
    float* __restrict__ yr, float* __restrict__ yi) {
    int k = blockIdx.x, b = blockIdx.y, i = threadIdx.x;
    __shared__ float sXr[DD], sXi[DD];
    sXr[i] = Xr[(b * KM + k) * DD + i];
    sXi[i] = Xi[(b * KM + k) * DD + i];
    __syncthreads();

    const float4* krow = (const float4*)(kr + (size_t)b * KD + ((size_t)k * DD + i) * DD);
    const float4* crow = (const float4*)(kc + (size_t)b * KD + ((size_t)k * DD + i) * DD);
    float aR = 0.f, aI = 0.f;
#pragma unroll
    for (int m4 = 0; m4 < DD / 4; ++m4) {
        float4 kv = krow[m4];
        float4 cv = crow[m4];
        int m = m4 * 4;
        aR += kv.x * sXr[m]   - cv.x * sXi[m];
        aI += kv.x * sXi[m]   + cv.x * sXr[m];
        aR += kv.y * sXr[m+1] - cv.y * sXi[m+1];
        aI += kv.y * sXi[m+1] + cv.y * sXr[m+1];
        aR += kv.z * sXr[m+2] - cv.z * sXi[m+2];
        aI += kv.z * sXi[m+2] + cv.z * sXr[m+2];
        aR += kv.w * sXr[m+3] - cv.w * sXi[m+3];
        aI += kv.w * sXi[m+3] + cv.w * sXr[m+3];
    }
    yr[(b * KM + k) * DD + i] = aR * (float)DD;   // einsum * d
    yi[(b * KM + k) * DD + i] = aI * (float)DD;
}

// ---------------------------------------------------------------------------
// Kernel 6: inverse DFT (only 32 non-zero modes), real part, * 1/N.
//   out[b,n,d] = (1/N) * sum_k [ yr*cos(2pi k n/N) - yi*sin(2pi k n/N) ]
// block = (64 d, 4 n); y rows are 256B coalesced + L1-reused across n.
// ---------------------------------------------------------------------------
__global__ __launch_bounds__(256) void idft_kernel(
    const float* __restrict__ yr, const float* __restrict__ yi,
    const float* __restrict__ cosT, const float* __restrict__ sinT,
    float* __restrict__ out) {
    int d = threadIdx.x;
    int n = blockIdx.x * 4 + threadIdx.y;
    int b = blockIdx.y;
    float acc = 0.f;
#pragma unroll
    for (int k = 0; k < KM; ++k) {
        float re = yr[(b * KM + k) * DD + d];
        float im = yi[(b * KM + k) * DD + d];
        acc = fmaf(re, cosT[n * KM + k], acc);
        acc = fmaf(-im, sinT[n * KM + k], acc);
    }
    out[((size_t)b * NN + n) * DD + d] = acc * (1.0f / (float)NN);
}

// ---------------------------------------------------------------------------
// Workspace layout (floats), total ~18.4 MB
// ---------------------------------------------------------------------------
#define WS_COS  ((size_t)0)
#define WS_SIN  (WS_COS + (size_t)NN * KM)          // 131072
#define WS_HR   (WS_SIN + (size_t)NN * KM)          // 262144
#define WS_HC   (WS_HR  + (size_t)BB * HH)          // +2048
#define WS_XR   (WS_HC  + (size_t)BB * HH)
#define WS_XI   (WS_XR  + (size_t)BB * KM * DD)     // +32768
#define WS_YR   (WS_XI  + (size_t)BB * KM * DD)
#define WS_YI   (WS_YR  + (size_t)BB * KM * DD)
#define WS_KR   (WS_YI  + (size_t)BB * KM * DD)
#define WS_KC   (WS_KR  + (size_t)BB * KD)          // +2097152
// end = WS_KC + BB*KD = 4,591,616 floats = 18.37 MB

extern "C" void kernel_launch(void* const* d_in, const int* in_sizes, int n_in,
                              void* d_out, int out_size, void* d_ws, size_t ws_size,
                              hipStream_t stream) {
    const float* params = (const float*)d_in[0];
    const float* f      = (const float*)d_in[1];
    const float* W1r    = (const float*)d_in[2];
    const float* b1r    = (const float*)d_in[3];
    const float* W2r    = (const float*)d_in[4];
    const float* b2r    = (const float*)d_in[5];
    const float* W1c    = (const float*)d_in[6];
    const float* b1c    = (const float*)d_in[7];
    const float* W2c    = (const float*)d_in[8];
    const float* b2c    = (const float*)d_in[9];
    float* out = (float*)d_out;
    float* ws  = (float*)d_ws;

    float* cosT = ws + WS_COS;
    float* sinT = ws + WS_SIN;
    float* hr   = ws + WS_HR;
    float* hc   = ws + WS_HC;
    float* Xr   = ws + WS_XR;
    float* Xi   = ws + WS_XI;
    float* yr   = ws + WS_YR;
    float* yi   = ws + WS_YI;
    float* kr   = ws + WS_KR;
    float* kc   = ws + WS_KC;

    // 1) trig table (1 MB, reused by both transforms)
    trig_kernel<<<dim3((NN * KM) / 256), dim3(256), 0, stream>>>(cosT, sinT);
    // 2) truncated forward DFT
    dft_kernel<<<dim3(KM / 4, BB), dim3(64, 4), 0, stream>>>(f, cosT, sinT, Xr, Xi);
    // 3) MLP hidden (real + imag branches)
    mlp_kernel<<<dim3(BB, 2), dim3(HH), 0, stream>>>(params, W1r, b1r, W1c, b1c, hr, hc);
    // 4) the 134 MB-streaming GEMM via f32 WMMA (sets the runtime floor)
    gemm_wmma_kernel<<<dim3(KD / 64, 2), dim3(128), 0, stream>>>(
        hr, hc, W2r, b2r, W2c, b2c, kr, kc);
    // 5) per-mode complex matvec (* D)
    matvec_kernel<<<dim3(KM, BB), dim3(64), 0, stream>>>(kr, kc, Xr, Xi, yr, yi);
    // 6) inverse DFT real part (* 1/N)
    idft_kernel<<<dim3(NN / 4, BB), dim3(64, 4), 0, stream>>>(yr, yi, cosT, sinT, out);
}